// MambaMemoryBlock_75771813037148
// MI455X (gfx1250) — compile-verified
//
#include <hip/hip_runtime.h>
#include <hip/hip_bf16.h>
#include <math.h>

#define Bn      8
#define Sn      12
#define NNn     325
#define Dn      128
#define EDn     256
#define DSTATEn 8
#define DCONVn  4
#define DTRANKn 8
#define MSLOTSn 10
#define Ln      (Sn*NNn)      /* 3900 */
#define NTOK    (Bn*Ln)       /* 31200 */
#define TM      3             /* M-tiles (of 16 tokens) per block */
#define TOKB    (16*TM)       /* 48 tokens per block */
#define NBLK    (NTOK/TOKB)   /* 650 (exact) */

typedef float v2f __attribute__((ext_vector_type(2)));
typedef float v8f __attribute__((ext_vector_type(8)));

// ---------------- WMMA f32 16x16x4 (CDNA5 layouts, ISA §7.12.2) ------------
__device__ __forceinline__ v8f wmma4(v2f a, v2f b, v8f c) {
  return __builtin_amdgcn_wmma_f32_16x16x4_f32(false, a, false, b, (short)0, c,
                                               false, false);
}

// C/D 16x16: VGPR j of lane l = D[j + 8*(l/16)][l%16]
__device__ __forceinline__ void store_c(float* C, int ldc, int nb, v8f c) {
  int lane = threadIdx.x & 31;
  int n = nb + (lane & 15);
  int mb = (lane < 16) ? 0 : 8;
#pragma unroll
  for (int j = 0; j < 8; ++j) C[(mb + j)*ldc + n] = c[j];
}

// GEMM inner loop: 3 M-tiles (rows t*16+m of A in LDS) x one N-tile,
// B pre-swizzled in fragment order: Bsw[(nt*KT + kt)*32 + lane] = {B[k][n],B[k+1][n]}
// Bounded unroll so A-fragments stay as in-loop ds_load_b64 (no spills).
__device__ __forceinline__ void gemm3(const float* A, int lda, const v2f* Bsw,
                                      int KT, int nt, v8f acc[TM]) {
  int lane = threadIdx.x & 31;
  int m = lane & 15;
  int ks = (lane >> 4) << 1;
  const v2f* bp = Bsw + (size_t)nt * KT * 32 + lane;
  const v2f* a0 = (const v2f*)(A + (0*16 + m)*lda + ks);
  const v2f* a1 = (const v2f*)(A + (1*16 + m)*lda + ks);
  const v2f* a2 = (const v2f*)(A + (2*16 + m)*lda + ks);
  v8f z = {};
  acc[0] = z; acc[1] = z; acc[2] = z;
#pragma unroll 4
  for (int kt = 0; kt < KT; ++kt) {      // k advances 4 floats = 2 v2f per kt
    v2f b = bp[kt * 32];
    acc[0] = wmma4(a0[kt*2], b, acc[0]);
    acc[1] = wmma4(a1[kt*2], b, acc[1]);
    acc[2] = wmma4(a2[kt*2], b, acc[2]);
  }
}

__device__ __forceinline__ float silu_f(float x) { return x / (1.f + __expf(-x)); }

// ---------------- weight swizzle: W[K][Nreal] -> fragment order ------------
__global__ void swz_kernel(const float* W, float* dst, int K, int Nreal, int NT) {
  int idx = blockIdx.x * blockDim.x + threadIdx.x;
  int KT = K / 4;
  if (idx >= NT * KT * 32) return;
  int lane = idx & 31;
  int kt = (idx >> 5) % KT;
  int nt = idx / (32 * KT);
  int n = nt*16 + (lane & 15);
  int k = kt*4 + ((lane >> 4) << 1);
  float bx = (n < Nreal) ? W[k*Nreal + n]     : 0.f;
  float by = (n < Nreal) ? W[(k+1)*Nreal + n] : 0.f;
  dst[idx*2]     = bx;
  dst[idx*2 + 1] = by;
}

// ---------------- K0: k = mem@Wk, v = mem@Wv (tiny) ------------------------
__global__ void kv_kernel(const float* mem, const float* Wk, const float* Wv,
                          float* kb, float* vb) {
  int i = blockIdx.x * blockDim.x + threadIdx.x;
  if (i >= 2 * MSLOTSn * Dn) return;
  int which = i / (MSLOTSn * Dn);
  int r = i % (MSLOTSn * Dn);
  int m = r / Dn, d = r % Dn;
  const float* W = which ? Wv : Wk;
  float acc = 0.f;
  for (int k = 0; k < Dn; ++k) acc += mem[m*Dn + k] * W[k*Dn + d];
  (which ? vb : kb)[m*Dn + d] = acc;
}

// k^T panel (KT=32, NT=1, N padded 10->16) and v panel (KT=4, NT=8, K pad 10->16)
__global__ void kvswz_kernel(const float* kb, const float* vb,
                             float* kfragT, float* vfrag) {
  int idx = blockIdx.x * blockDim.x + threadIdx.x;
  if (idx < 32*32) {
    int lane = idx & 31; int kt = idx >> 5;
    int n = lane & 15;   int k = kt*4 + ((lane >> 4) << 1);
    kfragT[idx*2]     = (n < MSLOTSn) ? kb[n*Dn + k]     : 0.f;
    kfragT[idx*2 + 1] = (n < MSLOTSn) ? kb[n*Dn + k + 1] : 0.f;
  }
  if (idx < 8*4*32) {
    int lane = idx & 31; int kt = (idx >> 5) & 3; int nt = idx >> 7;
    int n = nt*16 + (lane & 15); int k = kt*4 + ((lane >> 4) << 1);
    vfrag[idx*2]     = (k     < MSLOTSn) ? vb[k*Dn + n]     : 0.f;
    vfrag[idx*2 + 1] = (k + 1 < MSLOTSn) ? vb[(k+1)*Dn + n] : 0.f;
  }
}

// ---------------- K1: attention + Wo + LayerNorm + residual -> xr ----------
__global__ void attn_kernel(const float* x, const v2f* Wq_sw, const v2f* Wo_sw,
                            const float* ln_g, const float* ln_b,
                            const v2f* kfragT, const v2f* vfrag, float* xr) {
  __shared__ float xs[TOKB*Dn];
  __shared__ float qs[TOKB*Dn];
  __shared__ float av[TOKB*Dn];
  __shared__ float sc[TOKB*16];
  int tid = threadIdx.x;
  int wave = tid >> 5;
  long t0 = (long)blockIdx.x * TOKB;
  v8f acc[TM];

  for (int i = tid; i < TOKB*Dn; i += 128) xs[i] = x[t0*Dn + i];
  __syncthreads();

  // q = xs @ Wq  (NT=8 -> 2 per wave)
  for (int nti = 0; nti < 2; ++nti) {
    int nt = wave*2 + nti;
    gemm3(xs, Dn, Wq_sw, 32, nt, acc);
#pragma unroll
    for (int t = 0; t < TM; ++t) store_c(qs + t*16*Dn, Dn, nt*16, acc[t]);
  }
  __syncthreads();

  // scores = q @ k^T (single N-tile; wave 0)
  if (wave == 0) {
    gemm3(qs, Dn, kfragT, 32, 0, acc);
#pragma unroll
    for (int t = 0; t < TM; ++t) store_c(sc + t*16*16, 16, 0, acc[t]);
  }
  __syncthreads();

  // softmax over 10 slots per row
  if (tid < TOKB) {
    const float scale = 0.088388347648318447f;  // 1/sqrt(128)
    float mx = -1e30f;
    for (int m = 0; m < MSLOTSn; ++m) {
      float v = sc[tid*16 + m] * scale; sc[tid*16 + m] = v;
      if (v > mx) mx = v;
    }
    float s = 0.f;
    for (int m = 0; m < MSLOTSn; ++m) {
      float e = __expf(sc[tid*16 + m] - mx); sc[tid*16 + m] = e; s += e;
    }
    float inv = 1.f / s;
    for (int m = 0; m < MSLOTSn; ++m) sc[tid*16 + m] *= inv;
    for (int m = MSLOTSn; m < 16; ++m) sc[tid*16 + m] = 0.f;
  }
  __syncthreads();

  // av = att @ v  (NT=8 -> 2 per wave, KT=4)
  for (int nti = 0; nti < 2; ++nti) {
    int nt = wave*2 + nti;
    gemm3(sc, 16, vfrag, 4, nt, acc);
#pragma unroll
    for (int t = 0; t < TM; ++t) store_c(av + t*16*Dn, Dn, nt*16, acc[t]);
  }
  __syncthreads();

  // xo = av @ Wo -> qs (reuse)
  for (int nti = 0; nti < 2; ++nti) {
    int nt = wave*2 + nti;
    gemm3(av, Dn, Wo_sw, 32, nt, acc);
#pragma unroll
    for (int t = 0; t < TM; ++t) store_c(qs + t*16*Dn, Dn, nt*16, acc[t]);
  }
  __syncthreads();

  // LayerNorm + residual -> xr
  if (tid < TOKB) {
    float m = 0.f;
    for (int c = 0; c < Dn; ++c) m += qs[tid*Dn + c];
    m *= (1.f / Dn);
    float v = 0.f;
    for (int c = 0; c < Dn; ++c) { float d = qs[tid*Dn + c] - m; v += d*d; }
    v *= (1.f / Dn);
    float inv = rsqrtf(v + 1e-5f);
    long tok = t0 + tid;
    for (int c = 0; c < Dn; ++c)
      xr[tok*Dn + c] = (qs[tid*Dn + c] - m) * inv * ln_g[c] + ln_b[c]
                       + xs[tid*Dn + c];
  }
}

// ---------------- K2: RMSNorm + W_in (512 wide) -> xp, z -------------------
__global__ void in_kernel(const float* xr, const float* rms_w, const v2f* Win_sw,
                          float* xp, float* zb) {
  __shared__ float xn[TOKB*Dn];
  int tid = threadIdx.x;
  int wave = tid >> 5;
  int lane = tid & 31;
  long t0 = (long)blockIdx.x * TOKB;
  v8f acc[TM];

  for (int i = tid; i < TOKB*Dn; i += 128) xn[i] = xr[t0*Dn + i];
  __syncthreads();
  if (tid < TOKB) {
    float s = 0.f;
    for (int c = 0; c < Dn; ++c) { float v = xn[tid*Dn + c]; s += v*v; }
    float inv = rsqrtf(s / Dn + 1e-5f);
    for (int c = 0; c < Dn; ++c) xn[tid*Dn + c] = xn[tid*Dn + c] * inv * rms_w[c];
  }
  __syncthreads();

  // NT=32 -> 8 per wave
  for (int nti = 0; nti < 8; ++nti) {
    int nt = wave*8 + nti;
    gemm3(xn, Dn, Win_sw, 32, nt, acc);
    int n = nt*16 + (lane & 15);
    int mb = (lane < 16) ? 0 : 8;
#pragma unroll
    for (int t = 0; t < TM; ++t)
#pragma unroll
      for (int j = 0; j < 8; ++j) {
        long tok = t0 + t*16 + mb + j;
        float v = acc[t][j];
        if (n < EDn) xp[tok*EDn + n] = v;
        else         zb[tok*EDn + (n - EDn)] = v;
      }
  }
}

// ---------------- K3: causal depthwise conv + SiLU, W_x, delta, B, C -------
__global__ void conv_kernel(const float* xp, const float* conv_w, const float* conv_b,
                            const v2f* Wx_sw, const v2f* Wdt_sw, const float* b_dt,
                            float* xc, float* delta, float* Bm, float* Cm) {
  __shared__ float xcs[TOKB*EDn];
  __shared__ float dbcs[TOKB*32];
  int tid = threadIdx.x;
  int wave = tid >> 5;
  int lane = tid & 31;
  long t0 = (long)blockIdx.x * TOKB;
  v8f acc[TM];

  // conv (per-row; respects batch boundaries) + SiLU
  for (int i = tid; i < TOKB*EDn; i += 128) {
    int r = i >> 8;
    int c = i & 255;
    long tok = t0 + r;
    int b = (int)(tok / Ln);
    int l = (int)(tok % Ln);
    float a = conv_b[c];
#pragma unroll
    for (int j = 0; j < DCONVn; ++j) {
      int ll = l - (DCONVn - 1) + j;
      if (ll >= 0) a += conv_w[c*DCONVn + j] * xp[((long)b*Ln + ll)*EDn + c];
    }
    a = silu_f(a);
    xcs[i] = a;
    xc[tok*EDn + c] = a;
  }
  __syncthreads();

  // dbc = xc @ W_x  (N padded 24->32, NT=2: waves 0,1)
  if (wave < 2) {
    gemm3(xcs, EDn, Wx_sw, 64, wave, acc);
#pragma unroll
    for (int t = 0; t < TM; ++t) store_c(dbcs + t*16*32, 32, wave*16, acc[t]);
  }
  __syncthreads();

  // delta = softplus(dbc[:, :8] @ W_dt + b_dt)  (NT=16 -> 4 per wave, KT=2)
  for (int nti = 0; nti < 4; ++nti) {
    int nt = wave*4 + nti;
    gemm3(dbcs, 32, Wdt_sw, 2, nt, acc);
    int n = nt*16 + (lane & 15);
    int mb = (lane < 16) ? 0 : 8;
#pragma unroll
    for (int t = 0; t < TM; ++t)
#pragma unroll
      for (int j = 0; j < 8; ++j) {
        long tok = t0 + t*16 + mb + j;
        float v = acc[t][j] + b_dt[n];
        delta[tok*EDn + n] = (v > 20.f) ? v : log1pf(__expf(v));
      }
  }

  // split B, C
  for (int i = tid; i < TOKB*16; i += 128) {
    int r = i >> 4, c = i & 15;
    long tok = t0 + r;
    float v = dbcs[r*32 + DTRANKn + c];
    if (c < DSTATEn) Bm[tok*DSTATEn + c]             = v;
    else             Cm[tok*DSTATEn + (c - DSTATEn)] = v;
  }
}

// ---------------- K4: selective scan, one lane per (batch, channel) --------
__global__ void scan_kernel(const float* delta, const float* xc, const float* Bm,
                            const float* Cm, const float* A_log, const float* Dp,
                            float* y) {
  int gid = blockIdx.x * blockDim.x + threadIdx.x;
  if (gid >= Bn * EDn) return;
  int b = gid / EDn;
  int d = gid % EDn;
  float a[DSTATEn];
#pragma unroll
  for (int n = 0; n < DSTATEn; ++n) a[n] = -__expf(A_log[d*DSTATEn + n]);
  float dp = Dp[d];
  float h[DSTATEn];
#pragma unroll
  for (int n = 0; n < DSTATEn; ++n) h[n] = 0.f;
  long base = (long)b * Ln;
  for (int l = 0; l < Ln; ++l) {
    long t = base + l;
    float dt = delta[t*EDn + d];
    float xv = xc[t*EDn + d];
    float yv = 0.f;
#pragma unroll
    for (int n = 0; n < DSTATEn; ++n) {
      float dA = __expf(dt * a[n]);
      h[n] = dA * h[n] + dt * Bm[t*DSTATEn + n] * xv;
      yv += h[n] * Cm[t*DSTATEn + n];
    }
    y[t*EDn + d] = yv + dp * xv;
  }
}

// ---------------- K5: gate, W_out, residuals -> out ------------------------
__global__ void out_kernel(const float* y, const float* zb, const v2f* Wout_sw,
                           const float* xr, const float* x, float* out) {
  __shared__ float ys[TOKB*EDn];
  int tid = threadIdx.x;
  int wave = tid >> 5;
  int lane = tid & 31;
  long t0 = (long)blockIdx.x * TOKB;
  v8f acc[TM];

  for (int i = tid; i < TOKB*EDn; i += 128) {
    int r = i >> 8, c = i & 255;
    long tok = t0 + r;
    float z = zb[tok*EDn + c];
    ys[i] = y[tok*EDn + c] * silu_f(z);
  }
  __syncthreads();

  // NT=8 -> 2 per wave, KT=64
  for (int nti = 0; nti < 2; ++nti) {
    int nt = wave*2 + nti;
    gemm3(ys, EDn, Wout_sw, 64, nt, acc);
    int n = nt*16 + (lane & 15);
    int mb = (lane < 16) ? 0 : 8;
#pragma unroll
    for (int t = 0; t < TM; ++t)
#pragma unroll
      for (int j = 0; j < 8; ++j) {
        long tok = t0 + t*16 + mb + j;
        out[tok*Dn + n] = acc[t][j] + xr[tok*Dn + n] + x[tok*Dn + n];
      }
  }
}

// ---------------- host -----------------------------------------------------
extern "C" void kernel_launch(void* const* d_in, const int* in_sizes, int n_in,
                              void* d_out, int out_size, void* d_ws, size_t ws_size,
                              hipStream_t stream) {
  (void)in_sizes; (void)n_in; (void)out_size; (void)ws_size;
  const float* x      = (const float*)d_in[0];
  const float* mem    = (const float*)d_in[1];
  const float* Wq     = (const float*)d_in[2];
  const float* Wk     = (const float*)d_in[3];
  const float* Wv     = (const float*)d_in[4];
  const float* Wo     = (const float*)d_in[5];
  const float* ln1_g  = (const float*)d_in[6];
  const float* ln1_b  = (const float*)d_in[7];
  const float* rms_w  = (const float*)d_in[8];
  const float* W_in   = (const float*)d_in[9];
  const float* conv_w = (const float*)d_in[10];
  const float* conv_b = (const float*)d_in[11];
  const float* W_x    = (const float*)d_in[12];
  const float* W_dt   = (const float*)d_in[13];
  const float* b_dt   = (const float*)d_in[14];
  const float* A_log  = (const float*)d_in[15];
  const float* Dp     = (const float*)d_in[16];
  const float* W_out  = (const float*)d_in[17];

  float* ws = (float*)d_ws;
  size_t off = 0;
  float* kb      = ws + off; off += (size_t)MSLOTSn*Dn;
  float* vb      = ws + off; off += (size_t)MSLOTSn*Dn;
  float* Wq_sw   = ws + off; off += (size_t)8*32*32*2;
  float* Wo_sw   = ws + off; off += (size_t)8*32*32*2;
  float* Win_sw  = ws + off; off += (size_t)32*32*32*2;
  float* Wx_sw   = ws + off; off += (size_t)2*64*32*2;
  float* Wdt_sw  = ws + off; off += (size_t)16*2*32*2;
  float* Wout_sw = ws + off; off += (size_t)8*64*32*2;
  float* kfragT  = ws + off; off += (size_t)1*32*32*2;
  float* vfrag   = ws + off; off += (size_t)8*4*32*2;
  float* xr      = ws + off; off += (size_t)NTOK*Dn;
  float* xp      = ws + off; off += (size_t)NTOK*EDn;
  float* zb      = ws + off; off += (size_t)NTOK*EDn;
  float* xc      = ws + off; off += (size_t)NTOK*EDn;
  float* delta   = ws + off; off += (size_t)NTOK*EDn;
  float* Bm      = ws + off; off += (size_t)NTOK*DSTATEn;
  float* Cm      = ws + off; off += (size_t)NTOK*DSTATEn;
  float* y       = xp;  // xp dead after conv_kernel; reuse for scan output

  // weight swizzles (fragment order)
  swz_kernel<<<(8*32*32  + 255)/256, 256, 0, stream>>>(Wq,    Wq_sw,   Dn,  Dn,     8);
  swz_kernel<<<(8*32*32  + 255)/256, 256, 0, stream>>>(Wo,    Wo_sw,   Dn,  Dn,     8);
  swz_kernel<<<(32*32*32 + 255)/256, 256, 0, stream>>>(W_in,  Win_sw,  Dn,  2*EDn, 32);
  swz_kernel<<<(2*64*32  + 255)/256, 256, 0, stream>>>(W_x,   Wx_sw,   EDn, 24,     2);
  swz_kernel<<<(16*2*32  + 255)/256, 256, 0, stream>>>(W_dt,  Wdt_sw,  DTRANKn, EDn, 16);
  swz_kernel<<<(8*64*32  + 255)/256, 256, 0, stream>>>(W_out, Wout_sw, EDn, Dn,     8);

  kv_kernel   <<<(2*MSLOTSn*Dn + 255)/256, 256, 0, stream>>>(mem, Wk, Wv, kb, vb);
  kvswz_kernel<<<4, 256, 0, stream>>>(kb, vb, kfragT, vfrag);

  attn_kernel<<<NBLK, 128, 0, stream>>>(x, (const v2f*)Wq_sw, (const v2f*)Wo_sw,
                                        ln1_g, ln1_b, (const v2f*)kfragT,
                                        (const v2f*)vfrag, xr);
  in_kernel  <<<NBLK, 128, 0, stream>>>(xr, rms_w, (const v2f*)Win_sw, xp, zb);
  conv_kernel<<<NBLK, 128, 0, stream>>>(xp, conv_w, conv_b, (const v2f*)Wx_sw,
                                        (const v2f*)Wdt_sw, b_dt, xc, delta, Bm, Cm);
  scan_kernel<<<(Bn*EDn + 255)/256, 256, 0, stream>>>(delta, xc, Bm, Cm,
                                                      A_log, Dp, y);
  out_kernel <<<NBLK, 128, 0, stream>>>(y, zb, (const v2f*)Wout_sw, xr, x,
                                        (float*)d_out);
}